// MOELayer_36782099923441
// MI455X (gfx1250) — compile-verified
//
#include <hip/hip_runtime.h>
#include <hip/hip_bf16.h>
#include <math.h>

typedef __attribute__((ext_vector_type(2))) float v2f;
typedef __attribute__((ext_vector_type(8))) float v8f;

#define T_TOK 4096
#define DIM   1024
#define NEXP  8
#define CAP   640   // int(1.25 * 4096 / 8)
#define MTILE 32    // M rows per block (2 WMMA A-fragments)

// ---------------- ws layout (element offsets, 4B elems) ----------------
// top_idx  : int  [4096]      @ 0
// top_prob : float[4096]      @ 4096
// probs    : float[4096*8]    @ 8192
// f_cnt    : int  [8]         @ 40960
// cnt      : int  [8]         @ 40968
// list     : int  [8*640]     @ 40976

// ---------------------------------------------------------------- init
__global__ void moe_init_kernel(int* f_cnt) {
    if (threadIdx.x < NEXP) f_cnt[threadIdx.x] = 0;
}

// ------------------------------------------------------------- routing
__global__ __launch_bounds__(256)
void moe_route_kernel(const float* __restrict__ x,
                      const float* __restrict__ Wg,
                      const float* __restrict__ bg,
                      int* __restrict__ top_idx,
                      float* __restrict__ top_prob,
                      float* __restrict__ probs,
                      int* __restrict__ f_cnt) {
    const int wave = threadIdx.x >> 5;
    const int lane = threadIdx.x & 31;
    const int t = blockIdx.x * 8 + wave;

    float acc[NEXP];
#pragma unroll
    for (int e = 0; e < NEXP; ++e) acc[e] = 0.f;

    const float* xr = x + (size_t)t * DIM;
    for (int d = lane; d < DIM; d += 32) {
        const float xv = xr[d];
        const float* wr = Wg + (size_t)d * NEXP;
#pragma unroll
        for (int e = 0; e < NEXP; ++e) acc[e] += xv * wr[e];
    }
#pragma unroll
    for (int off = 16; off > 0; off >>= 1) {
#pragma unroll
        for (int e = 0; e < NEXP; ++e) acc[e] += __shfl_xor(acc[e], off, 32);
    }

    if (lane == 0) {
        float lg[NEXP];
        float mx = -INFINITY;
        int arg = 0;
#pragma unroll
        for (int e = 0; e < NEXP; ++e) {
            lg[e] = acc[e] + bg[e];
            if (lg[e] > mx) { mx = lg[e]; arg = e; }   // strict '>' keeps first max
        }
        float s = 0.f;
        float pe[NEXP];
#pragma unroll
        for (int e = 0; e < NEXP; ++e) { pe[e] = __expf(lg[e] - mx); s += pe[e]; }
        const float inv = 1.f / s;
#pragma unroll
        for (int e = 0; e < NEXP; ++e) probs[(size_t)t * NEXP + e] = pe[e] * inv;
        top_idx[t]  = arg;
        top_prob[t] = pe[arg] * inv;
        atomicAdd(&f_cnt[arg], 1);                      // int atomic: deterministic
    }
}

// ----------------------------------------------------- aux loss (det.)
__global__ __launch_bounds__(256)
void moe_aux_kernel(const float* __restrict__ probs,
                    const int* __restrict__ f_cnt,
                    float* __restrict__ out_aux) {
    const int wave = threadIdx.x >> 5;
    const int lane = threadIdx.x & 31;
    float s = 0.f;
    for (int t = lane; t < T_TOK; t += 32) s += probs[(size_t)t * NEXP + wave];
#pragma unroll
    for (int off = 16; off > 0; off >>= 1) s += __shfl_xor(s, off, 32);

    __shared__ float fp[NEXP];
    if (lane == 0) fp[wave] = s * (float)f_cnt[wave];
    __syncthreads();
    if (threadIdx.x == 0) {
        float a = 0.f;
#pragma unroll
        for (int e = 0; e < NEXP; ++e) a += fp[e];
        out_aux[0] = a * (float)NEXP / ((float)T_TOK * (float)T_TOK);
    }
}

// ---------------------------------------------------- scan + compaction
__global__ void moe_scan_kernel(const int* __restrict__ top_idx,
                                int* __restrict__ cnt,
                                int* __restrict__ list) {
    const int lane = threadIdx.x;
    const unsigned mask_lt = (1u << lane) - 1u;
    int run[NEXP];
#pragma unroll
    for (int e = 0; e < NEXP; ++e) run[e] = 0;

    for (int c = 0; c < T_TOK / 32; ++c) {
        const int t = c * 32 + lane;
        const int my = top_idx[t];
        int pos = -1;
#pragma unroll
        for (int e = 0; e < NEXP; ++e) {
            const unsigned m = (unsigned)__ballot(my == e);
            if (my == e) pos = run[e] + __popc(m & mask_lt);
            run[e] += __popc(m);
        }
        if (pos < CAP) list[my * CAP + pos] = t;
    }
    if (lane < NEXP) cnt[lane] = run[lane] < CAP ? run[lane] : CAP;
}

// ------------------------------------------------------------ zero out
__global__ __launch_bounds__(256)
void moe_zero_kernel(float4* __restrict__ out) {
    const size_t i = (size_t)blockIdx.x * 256 + threadIdx.x;
    out[i] = make_float4(0.f, 0.f, 0.f, 0.f);
}

// -------------------------------------------------------- expert GEMM
// Block: 128 threads = 4 waves. Wave computes 32(M) x 64(N) with 8
// accumulators; each B fragment feeds TWO WMMAs (A-frag rows 0-15 / 16-31),
// halving global B traffic per FLOP. Block covers 32 x 256.
// Grid = (4 Nblk, 20 Mtile, 8 expert).
__global__ __launch_bounds__(128)
void moe_gemm_kernel(const float* __restrict__ x,
                     const float* __restrict__ W,
                     const float* __restrict__ bias,
                     const int* __restrict__ cnt,
                     const int* __restrict__ list,
                     const float* __restrict__ top_prob,
                     float* __restrict__ out) {
    const int e  = blockIdx.z;
    const int mt = blockIdx.y;
    const int ce = cnt[e];
    if (mt * MTILE >= ce) return;

    // [m][k] with pitch 36: 144B row stride keeps b64/b128 alignment and
    // gives conflict-free banks (9*m mod 16 is a permutation).
    __shared__ __align__(16) float Xs[MTILE][36];
    __shared__ int tokS[MTILE];

    const int tid  = threadIdx.x;
    const int lane = tid & 31;
    const int wave = tid >> 5;

    if (tid < MTILE) {
        const int i = mt * MTILE + tid;
        tokS[tid] = (i < ce) ? list[e * CAP + i] : -1;
    }
    __syncthreads();

    const int m    = lane & 15;             // A-frag row / B,D column in tile
    const int klo  = (lane >> 4) << 1;      // lanes 0-15: K {0,1}; 16-31: K {2,3}
    const int nBase = blockIdx.x * 256 + wave * 64 + m;
    const float* Wp = W + (size_t)e * DIM * DIM;

    v8f acc0a = {}, acc1a = {}, acc2a = {}, acc3a = {};
    v8f acc0b = {}, acc1b = {}, acc2b = {}, acc3b = {};

    for (int kc = 0; kc < DIM; kc += 32) {
        __syncthreads();
        // Cooperative A-tile load: 32 rows x 32 k, float4-coalesced.
#pragma unroll
        for (int it = 0; it < 2; ++it) {
            const int q   = tid + it * 128;   // 0..255
            const int ml  = q >> 3;           // row 0..31
            const int k4  = (q & 7) << 2;     // 0,4,...,28
            const int tok = tokS[ml];
            float4 v = make_float4(0.f, 0.f, 0.f, 0.f);
            if (tok >= 0) v = *(const float4*)(x + (size_t)tok * DIM + kc + k4);
            *(float4*)&Xs[ml][k4] = v;
        }
        __syncthreads();

#pragma unroll
        for (int kk = 0; kk < 32; kk += 4) {
            const v2f a0 = *(const v2f*)&Xs[m][kk + klo];        // rows 0-15
            const v2f a1 = *(const v2f*)&Xs[m + 16][kk + klo];   // rows 16-31

            const float* wr = Wp + (size_t)(kc + kk + klo) * DIM;
            v2f b0, b1, b2, b3;
            b0.x = wr[nBase];       b0.y = wr[DIM + nBase];
            b1.x = wr[nBase + 16];  b1.y = wr[DIM + nBase + 16];
            b2.x = wr[nBase + 32];  b2.y = wr[DIM + nBase + 32];
            b3.x = wr[nBase + 48];  b3.y = wr[DIM + nBase + 48];

            acc0a = __builtin_amdgcn_wmma_f32_16x16x4_f32(false, a0, false, b0,
                                                          (short)0, acc0a, false, false);
            acc0b = __builtin_amdgcn_wmma_f32_16x16x4_f32(false, a1, false, b0,
                                                          (short)0, acc0b, false, false);
            acc1a = __builtin_amdgcn_wmma_f32_16x16x4_f32(false, a0, false, b1,
                                                          (short)0, acc1a, false, false);
            acc1b = __builtin_amdgcn_wmma_f32_16x16x4_f32(false, a1, false, b1,
                                                          (short)0, acc1b, false, false);
            acc2a = __builtin_amdgcn_wmma_f32_16x16x4_f32(false, a0, false, b2,
                                                          (short)0, acc2a, false, false);
            acc2b = __builtin_amdgcn_wmma_f32_16x16x4_f32(false, a1, false, b2,
                                                          (short)0, acc2b, false, false);
            acc3a = __builtin_amdgcn_wmma_f32_16x16x4_f32(false, a0, false, b3,
                                                          (short)0, acc3a, false, false);
            acc3b = __builtin_amdgcn_wmma_f32_16x16x4_f32(false, a1, false, b3,
                                                          (short)0, acc3b, false, false);
        }
    }

    // Epilogue: D layout -> VGPR r holds rows r (lanes 0-15) / r+8 (lanes 16-31).
    const int mrow = (lane >> 4) << 3;
    const float* be = bias + (size_t)e * DIM;
    const float b0 = be[nBase], b1 = be[nBase + 16], b2 = be[nBase + 32], b3 = be[nBase + 48];

#pragma unroll
    for (int r = 0; r < 8; ++r) {
        // first A-fragment: rows 0-15
        int ml  = mrow + r;
        int tok = tokS[ml];
        if (tok >= 0) {
            const float p = top_prob[tok];
            float* orow = out + (size_t)tok * DIM;
            orow[nBase]      = (acc0a[r] + b0) * p;
            orow[nBase + 16] = (acc1a[r] + b1) * p;
            orow[nBase + 32] = (acc2a[r] + b2) * p;
            orow[nBase + 48] = (acc3a[r] + b3) * p;
        }
        // second A-fragment: rows 16-31
        ml  = 16 + mrow + r;
        tok = tokS[ml];
        if (tok >= 0) {
            const float p = top_prob[tok];
            float* orow = out + (size_t)tok * DIM;
            orow[nBase]      = (acc0b[r] + b0) * p;
            orow[nBase + 16] = (acc1b[r] + b1) * p;
            orow[nBase + 32] = (acc2b[r] + b2) * p;
            orow[nBase + 48] = (acc3b[r] + b3) * p;
        }
    }
}

// ---------------------------------------------------------------- host
extern "C" void kernel_launch(void* const* d_in, const int* in_sizes, int n_in,
                              void* d_out, int out_size, void* d_ws, size_t ws_size,
                              hipStream_t stream) {
    const float* x  = (const float*)d_in[0];   // [4096,1024]
    const float* Wg = (const float*)d_in[1];   // [1024,8]
    const float* bg = (const float*)d_in[2];   // [8]
    const float* W  = (const float*)d_in[3];   // [8,1024,1024]
    const float* b  = (const float*)d_in[4];   // [8,1024]
    float* out = (float*)d_out;                // [4096*1024] + 1 (aux)

    int*   top_idx  = (int*)d_ws;
    float* top_prob = (float*)d_ws + 4096;
    float* probs    = (float*)d_ws + 8192;
    int*   f_cnt    = (int*)d_ws + 40960;
    int*   cnt      = (int*)d_ws + 40968;
    int*   list     = (int*)d_ws + 40976;

    moe_init_kernel<<<1, 32, 0, stream>>>(f_cnt);

    moe_route_kernel<<<T_TOK / 8, 256, 0, stream>>>(x, Wg, bg, top_idx, top_prob,
                                                    probs, f_cnt);

    moe_aux_kernel<<<1, 256, 0, stream>>>(probs, f_cnt, out + (size_t)T_TOK * DIM);

    moe_scan_kernel<<<1, 32, 0, stream>>>(top_idx, cnt, list);

    moe_zero_kernel<<<(T_TOK * DIM) / (4 * 256), 256, 0, stream>>>((float4*)out);

    dim3 grid(DIM / 256, CAP / MTILE, NEXP);   // (4, 20, 8)
    moe_gemm_kernel<<<grid, 128, 0, stream>>>(x, W, b, cnt, list, top_prob, out);
}